// GIN_41540923686985
// MI455X (gfx1250) — compile-verified
//
#include <hip/hip_runtime.h>
#include <hip/hip_bf16.h>
#include <math.h>

#define NODES   50000
#define EDGES   600000
#define DCH     128
#define GRAPHS  500
#define OUTC    10
#define TILES   (NODES / 16)   // 3125, exact

typedef __attribute__((ext_vector_type(2))) float v2f;
typedef __attribute__((ext_vector_type(8))) float v8f;

// ---------------------------------------------------------------- zero fill
__global__ void gin_zero(float* __restrict__ p, int n) {
    int i = blockIdx.x * blockDim.x + threadIdx.x;
    if (i < n) p[i] = 0.0f;
}

// ------------------------------------------------- edge scatter: agg[dst]+=h[src]
// One thread handles (edge, 32-channel chunk). float4 gathers, f32 atomics.
__global__ void gin_scatter(const float* __restrict__ h,
                            const int* __restrict__ src,
                            const int* __restrict__ dst,
                            float* __restrict__ agg) {
    int idx = blockIdx.x * blockDim.x + threadIdx.x;
    int e = idx >> 2;
    if (e >= EDGES) return;
    int c0 = (idx & 3) * 32;
    int s = src[e];
    int d = dst[e];
    const float4* in = (const float4*)(h + (size_t)s * DCH + c0);
    float* out = agg + (size_t)d * DCH + c0;
#pragma unroll
    for (int q = 0; q < 8; ++q) {
        float4 v = in[q];
        atomicAdd(out + q * 4 + 0, v.x);
        atomicAdd(out + q * 4 + 1, v.y);
        atomicAdd(out + q * 4 + 2, v.z);
        atomicAdd(out + q * 4 + 3, v.w);
    }
}

// ------------------------------------------------- GIN MLP on one 16-node tile
// z = h + agg ; t = relu(z@W1 + b1) ; out = t@W2 + b2 (+relu if apply_relu)
// Single wave32 per block. WMMA f32 16x16x4; 8 accumulators cover N=128.
__global__ __launch_bounds__(32) void gin_mlp_wmma(
    const float* __restrict__ h_in, const float* __restrict__ agg,
    const float* __restrict__ W1, const float* __restrict__ b1,
    const float* __restrict__ W2, const float* __restrict__ b2,
    float* __restrict__ h_out, int apply_relu)
{
    __shared__ float lds_t[16 * DCH];     // 8 KB intermediate (C/D -> A relayout)

    const int lane  = threadIdx.x;        // 0..31
    const int m     = lane & 15;          // row within tile (A/B lane row)
    const int kb    = (lane >> 4) * 2;    // 32-bit A layout: lanes>=16 hold K=2,3
    const int half  = lane >> 4;          // 0 or 1 -> C/D rows r / r+8
    const int node0 = blockIdx.x * 16;

    // ---------------- GEMM 1: A = (h + agg) row m, B = W1 ----------------
    v8f acc[8];
#pragma unroll
    for (int nt = 0; nt < 8; ++nt)
        acc[nt] = (v8f){0.f,0.f,0.f,0.f,0.f,0.f,0.f,0.f};

    const float* hrow = h_in + (size_t)(node0 + m) * DCH;
    const float* arow = agg  + (size_t)(node0 + m) * DCH;

    for (int k0 = 0; k0 < DCH; k0 += 4) {
        v2f a;
        a.x = hrow[k0 + kb]     + arow[k0 + kb];
        a.y = hrow[k0 + kb + 1] + arow[k0 + kb + 1];
        const float* w0 = W1 + (size_t)(k0 + kb)     * DCH;
        const float* w1 = W1 + (size_t)(k0 + kb + 1) * DCH;
#pragma unroll
        for (int nt = 0; nt < 8; ++nt) {
            int col = nt * 16 + m;
            v2f b;
            b.x = w0[col];
            b.y = w1[col];
            acc[nt] = __builtin_amdgcn_wmma_f32_16x16x4_f32(
                false, a, false, b, (short)0, acc[nt], false, false);
        }
    }

    // bias + relu, spill t (16x128) to LDS in row-major for the 2nd GEMM
#pragma unroll
    for (int nt = 0; nt < 8; ++nt) {
        int col = nt * 16 + m;
        float bv = b1[col];
#pragma unroll
        for (int r = 0; r < 8; ++r) {
            float v = acc[nt][r] + bv;
            v = fmaxf(v, 0.0f);
            int trow = r + 8 * half;      // C/D layout: lanes>=16 hold rows 8..15
            lds_t[trow * DCH + col] = v;
        }
    }
    __syncthreads();                      // 1-wave block: ordering fence only

    // ---------------- GEMM 2: A = t (from LDS), B = W2 ----------------
#pragma unroll
    for (int nt = 0; nt < 8; ++nt)
        acc[nt] = (v8f){0.f,0.f,0.f,0.f,0.f,0.f,0.f,0.f};

    for (int k0 = 0; k0 < DCH; k0 += 4) {
        v2f a;
        a.x = lds_t[m * DCH + k0 + kb];
        a.y = lds_t[m * DCH + k0 + kb + 1];
        const float* w0 = W2 + (size_t)(k0 + kb)     * DCH;
        const float* w1 = W2 + (size_t)(k0 + kb + 1) * DCH;
#pragma unroll
        for (int nt = 0; nt < 8; ++nt) {
            int col = nt * 16 + m;
            v2f b;
            b.x = w0[col];
            b.y = w1[col];
            acc[nt] = __builtin_amdgcn_wmma_f32_16x16x4_f32(
                false, a, false, b, (short)0, acc[nt], false, false);
        }
    }

    // bias (+relu) and store to global
#pragma unroll
    for (int nt = 0; nt < 8; ++nt) {
        int col = nt * 16 + m;
        float bv = b2[col];
#pragma unroll
        for (int r = 0; r < 8; ++r) {
            float v = acc[nt][r] + bv;
            if (apply_relu) v = fmaxf(v, 0.0f);
            int trow = r + 8 * half;
            h_out[(size_t)(node0 + trow) * DCH + col] = v;
        }
    }
}

// ------------------------------------------------- global_add_pool into d_out
__global__ void gin_pool(const float* __restrict__ h,
                         const int* __restrict__ batch,
                         float* __restrict__ hG) {
    int idx = blockIdx.x * blockDim.x + threadIdx.x;
    int n = idx >> 2;
    if (n >= NODES) return;
    int c0 = (idx & 3) * 32;
    int g = batch[n];
    const float4* in = (const float4*)(h + (size_t)n * DCH + c0);
    float* out = hG + (size_t)g * DCH + c0;
#pragma unroll
    for (int q = 0; q < 8; ++q) {
        float4 v = in[q];
        atomicAdd(out + q * 4 + 0, v.x);
        atomicAdd(out + q * 4 + 1, v.y);
        atomicAdd(out + q * 4 + 2, v.z);
        atomicAdd(out + q * 4 + 3, v.w);
    }
}

// ------------------------------------------------- head: linear + log_softmax
__global__ void gin_head(const float* __restrict__ hG,
                         const float* __restrict__ lin_w,
                         const float* __restrict__ lin_b,
                         float* __restrict__ out_ls) {
    int g = blockIdx.x * blockDim.x + threadIdx.x;
    if (g >= GRAPHS) return;
    float logits[OUTC];
#pragma unroll
    for (int o = 0; o < OUTC; ++o) logits[o] = lin_b[o];
    for (int k = 0; k < DCH; ++k) {
        float hv = hG[(size_t)g * DCH + k];
#pragma unroll
        for (int o = 0; o < OUTC; ++o)
            logits[o] += hv * lin_w[k * OUTC + o];
    }
    float mx = logits[0];
#pragma unroll
    for (int o = 1; o < OUTC; ++o) mx = fmaxf(mx, logits[o]);
    float s = 0.0f;
#pragma unroll
    for (int o = 0; o < OUTC; ++o) s += expf(logits[o] - mx);
    float ls = logf(s);
#pragma unroll
    for (int o = 0; o < OUTC; ++o)
        out_ls[(size_t)g * OUTC + o] = logits[o] - mx - ls;
}

// ---------------------------------------------------------------- host launch
extern "C" void kernel_launch(void* const* d_in, const int* in_sizes, int n_in,
                              void* d_out, int out_size, void* d_ws, size_t ws_size,
                              hipStream_t stream) {
    const float* x     = (const float*)d_in[0];
    const int*   eidx  = (const int*)d_in[1];
    const int*   batch = (const int*)d_in[2];
    const float* W1    = (const float*)d_in[3];
    const float* b1    = (const float*)d_in[4];
    const float* W2    = (const float*)d_in[5];
    const float* b2    = (const float*)d_in[6];
    const float* lin_w = (const float*)d_in[7];
    const float* lin_b = (const float*)d_in[8];

    const int* src = eidx;
    const int* dst = eidx + EDGES;

    float* ws   = (float*)d_ws;
    float* hA   = ws;                         // NODES*DCH
    float* hB   = ws + (size_t)NODES * DCH;   // NODES*DCH
    float* agg  = ws + (size_t)2 * NODES * DCH;

    float* hG     = (float*)d_out;            // [GRAPHS, DCH]
    float* out_ls = (float*)d_out + (size_t)GRAPHS * DCH;  // [GRAPHS, OUTC]

    const int feat = NODES * DCH;             // 6,400,000

    const float* h_in = x;
    float* h_out = hA;
    for (int layer = 0; layer < 4; ++layer) {
        gin_zero<<<(feat + 255) / 256, 256, 0, stream>>>(agg, feat);
        gin_scatter<<<(EDGES * 4 + 255) / 256, 256, 0, stream>>>(h_in, src, dst, agg);
        gin_mlp_wmma<<<TILES, 32, 0, stream>>>(
            h_in, agg,
            W1 + (size_t)layer * DCH * DCH, b1 + (size_t)layer * DCH,
            W2 + (size_t)layer * DCH * DCH, b2 + (size_t)layer * DCH,
            h_out, (layer < 3) ? 1 : 0);
        h_in = h_out;
        h_out = (h_out == hA) ? hB : hA;
    }
    // h_in now points at the final node features

    gin_zero<<<(GRAPHS * DCH + 255) / 256, 256, 0, stream>>>(hG, GRAPHS * DCH);
    gin_pool<<<(NODES * 4 + 255) / 256, 256, 0, stream>>>(h_in, batch, hG);
    gin_head<<<(GRAPHS + 255) / 256, 256, 0, stream>>>(hG, lin_w, lin_b, out_ls);
}